// RetrievalHead_84112639525596
// MI455X (gfx1250) — compile-verified
//
#include <hip/hip_runtime.h>
#include <hip/hip_bf16.h>
#include <math.h>

// Problem constants (match reference)
#define B_     1024
#define N_     131072
#define DM_    2048
#define DC_    64
#define TOPK_  16

// Partitioning of the fused GEMM + top-k kernel
#define PPARTS   64              // N partitions (grid.x)
#define NPART    (N_ / PPARTS)   // 2048 candidates per partition
#define CHUNK    64              // stored rows staged in LDS per iteration
#define NCHUNKS  (NPART / CHUNK) // 32

typedef __attribute__((ext_vector_type(2))) float v2f;
typedef __attribute__((ext_vector_type(8))) float v8f;

// ---------------------------------------------------------------------------
// Kernel 1: G = M @ M^T   (64x64)
// ---------------------------------------------------------------------------
__global__ __launch_bounds__(256) void kMetric(const float* __restrict__ M,
                                               float* __restrict__ G) {
    int tid = threadIdx.x;
    for (int e = tid; e < DC_ * DC_; e += 256) {
        int i = e >> 6, j = e & 63;
        float s = 0.f;
        #pragma unroll 8
        for (int k = 0; k < DC_; ++k) s += M[i * DC_ + k] * M[j * DC_ + k];
        G[e] = s;
    }
}

// ---------------------------------------------------------------------------
// Kernel 2: per query row b: qc = query[b] @ W_proj ; qG = qc @ G ; q2 = qc.qG
// ---------------------------------------------------------------------------
__global__ __launch_bounds__(256) void kQuery(const float* __restrict__ q,
                                              const float* __restrict__ W,
                                              const float* __restrict__ G,
                                              float* __restrict__ qG,
                                              float* __restrict__ q2) {
    __shared__ float part[4][DC_];
    __shared__ float qc[DC_];
    __shared__ float pr[DC_];
    int b = blockIdx.x;
    int tid = threadIdx.x;
    int c = tid & 63, kg = tid >> 6;

    const float* qrow = q + (size_t)b * DM_;
    float s = 0.f;
    for (int i = 0; i < DM_ / 4; ++i) {
        int k = kg * (DM_ / 4) + i;
        s += qrow[k] * W[(size_t)k * DC_ + c];
    }
    part[kg][c] = s;
    __syncthreads();
    if (kg == 0) qc[c] = part[0][c] + part[1][c] + part[2][c] + part[3][c];
    __syncthreads();
    if (kg == 0) {
        float g = 0.f;
        #pragma unroll 8
        for (int j = 0; j < DC_; ++j) g += qc[j] * G[j * DC_ + c];
        qG[(size_t)b * DC_ + c] = g;
        pr[c] = g * qc[c];
    }
    __syncthreads();
    if (tid == 0) {
        float t = 0.f;
        for (int j = 0; j < DC_; ++j) t += pr[j];
        q2[b] = t;
    }
}

// ---------------------------------------------------------------------------
// Kernel 3: s2[n] = stored[n] . (G @ stored[n]) and rinv[n] = 1/(res[n]+eps)
// 256 blocks x 256 threads; 4 rows per iteration, G cached in LDS.
// ---------------------------------------------------------------------------
__global__ __launch_bounds__(256) void kS2(const float* __restrict__ S,
                                           const float* __restrict__ G,
                                           const float* __restrict__ res,
                                           float* __restrict__ s2,
                                           float* __restrict__ rinv) {
    __shared__ float Gs[DC_ * DC_];
    __shared__ float srow[4][DC_];
    __shared__ float pr[4][DC_];
    int tid = threadIdx.x;
    for (int e = tid; e < DC_ * DC_; e += 256) Gs[e] = G[e];
    int lr = tid >> 6, c = tid & 63;
    int base = blockIdx.x * (N_ / 256);          // 512 rows per block
    // full-precision reciprocal of weighted resilience (hot loop uses a mul)
    for (int n = base + tid; n < base + (N_ / 256); n += 256)
        rinv[n] = 1.0f / (res[n] + 1e-8f);
    __syncthreads();
    for (int it = 0; it < (N_ / 256) / 4; ++it) {
        int n = base + it * 4 + lr;
        float sv = S[(size_t)n * DC_ + c];
        srow[lr][c] = sv;
        __syncthreads();
        float g = 0.f;
        #pragma unroll 8
        for (int j = 0; j < DC_; ++j) g += srow[lr][j] * Gs[j * DC_ + c];
        pr[lr][c] = g * sv;
        __syncthreads();
        if (c == 0) {
            float t = 0.f;
            for (int j = 0; j < DC_; ++j) t += pr[lr][j];
            s2[n] = t;
        }
        __syncthreads();
    }
}

// ---------------------------------------------------------------------------
// Kernel 4: fused  cross = qG @ stored^T  (V_WMMA_F32_16X16X4_F32)
//           -> weighted geodesic distance -> per-partition top-16 per row.
// grid = (PPARTS, B/16), block = 128 (4 waves). Wave w owns tile columns
// w*16..w*16+15 of the 64-wide LDS chunk; all waves share the 16-query strip.
// A fragments + q2 hoisted to registers; B fragments bulk-prefetched per tile
// so the 16 DS loads pipeline ahead of the pure D->C WMMA accumulation chain.
// ---------------------------------------------------------------------------
__global__ __launch_bounds__(128) void kMain(const float* __restrict__ qG,
                                             const float* __restrict__ q2,
                                             const float* __restrict__ S,
                                             const float* __restrict__ s2,
                                             const float* __restrict__ rinv,
                                             float* __restrict__ pv,
                                             int* __restrict__ pix) {
    __shared__ float sQ[16 * 65];          // padded: conflict-free column reads
    __shared__ float q2s[16];
    __shared__ float sS[CHUNK * 65];       // padded stored chunk
    __shared__ float s2c[CHUNK];
    __shared__ float rc[CHUNK];
    __shared__ float sT[4][16 * 17];       // per-wave tile transpose buffer
    __shared__ float mv[16][132];          // merge: 128 candidates per row
    __shared__ int   mi_[16][132];

    const int tid = threadIdx.x;
    const int w   = tid >> 5;              // wave id 0..3 (wave32)
    const int l   = tid & 31;
    const int part = blockIdx.x;
    const int qb   = blockIdx.y * 16;

    // stage the 16-query qG strip + q2 (vectorized b128 loads)
    {
        const float4* src = (const float4*)(qG + (size_t)qb * DC_);
        for (int i = tid; i < 16 * DC_ / 4; i += 128) {
            float4 v = src[i];
            int e = i * 4, m = e >> 6, k = e & 63;
            float* dst = &sQ[m * 65 + k];
            dst[0] = v.x; dst[1] = v.y; dst[2] = v.z; dst[3] = v.w;
        }
        if (tid < 16) q2s[tid] = q2[qb + tid];
    }
    __syncthreads();

    const int m_  = l & 15;                // fragment row (A) / tile column (B,C)
    const int hi  = l >> 4;
    const int koff = hi * 2;               // K lane-half offset (K0/K1 vs K2/K3)

    // hoist loop-invariant A fragments (16 x v2f = 32 VGPRs) and q2 rows
    v2f afrag[16];
    {
        const float* aRow = &sQ[m_ * 65];
        #pragma unroll
        for (int kk = 0; kk < 16; ++kk) {
            int kb = kk * 4 + koff;
            afrag[kk].x = aRow[kb];
            afrag[kk].y = aRow[kb + 1];
        }
    }
    float q2r[8];
    #pragma unroll
    for (int r = 0; r < 8; ++r) q2r[r] = q2s[r + hi * 8];

    // per-lane running top-16 (replace-max trick)
    float lv[TOPK_]; int li[TOPK_];
    #pragma unroll
    for (int i = 0; i < TOPK_; ++i) { lv[i] = 3.0e38f; li[i] = 0x7fffffff; }
    float worst = 3.0e38f; int worstPos = 0;

    for (int chunk = 0; chunk < NCHUNKS; ++chunk) {
        const int nbase = part * NPART + chunk * CHUNK;
        __syncthreads();                   // previous chunk's sS reads done
        {
            const float4* src = (const float4*)(S + (size_t)nbase * DC_);
            for (int i = tid; i < CHUNK * DC_ / 4; i += 128) {
                float4 v = src[i];
                int e = i * 4, n = e >> 6, k = e & 63;
                float* dst = &sS[n * 65 + k];
                dst[0] = v.x; dst[1] = v.y; dst[2] = v.z; dst[3] = v.w;
            }
            if (tid < 64) s2c[tid]      = s2[nbase + tid];
            else          rc[tid - 64]  = rinv[nbase + (tid - 64)];
        }
        __syncthreads();

        // bulk-prefetch all 16 B fragments (pipelined ds_load_2addr_b32)
        v2f bfrag[16];
        {
            const float* bRow = &sS[(w * 16 + m_) * 65];
            #pragma unroll
            for (int kk = 0; kk < 16; ++kk) {
                int kb = kk * 4 + koff;
                bfrag[kk].x = bRow[kb];
                bfrag[kk].y = bRow[kb + 1];
            }
        }

        // 16x16 tile, K = 64: pure D->C accumulation chain (full-rate pattern)
        v8f acc = {};
        #pragma unroll
        for (int kk = 0; kk < 16; ++kk) {
            acc = __builtin_amdgcn_wmma_f32_16x16x4_f32(
                false, afrag[kk], false, bfrag[kk], (short)0, acc, false, false);
        }

        // cross -> distance, staged through per-wave LDS transpose
        float* T = sT[w];
        const int ncol = w * 16 + m_;      // chunk-local column of this lane
        const float s2v = s2c[ncol];
        const float rcv = rc[ncol];
        #pragma unroll
        for (int r = 0; r < 8; ++r) {
            int mrow = r + hi * 8;
            float d2 = q2r[r] + s2v - 2.0f * acc[r];
            float dist = sqrtf(fmaxf(d2, 0.0f) + 1e-12f) * rcv;
            T[mrow * 17 + m_] = dist;
        }
        // wave-local DS RAW: DS ops are in-order per wave; wait + compiler fence
        asm volatile("s_wait_dscnt 0" ::: "memory");

        // owner lanes (m_, m_+16) ingest 8 columns each for row m_
        #pragma unroll
        for (int j = 0; j < 8; ++j) {
            int colt = hi * 8 + j;
            float d = T[m_ * 17 + colt];
            int gidx = nbase + w * 16 + colt;
            if (d < worst) {
                #pragma unroll
                for (int i = 0; i < TOPK_; ++i)
                    if (i == worstPos) { lv[i] = d; li[i] = gidx; }
                worst = lv[0]; worstPos = 0;
                #pragma unroll
                for (int i = 1; i < TOPK_; ++i)
                    if (lv[i] > worst) { worst = lv[i]; worstPos = i; }
            }
        }
    }

    // merge the 8 owner-lane lists per row -> partition-local top-16
    __syncthreads();
    {
        int slot = w * 32 + hi * 16;
        #pragma unroll
        for (int i = 0; i < TOPK_; ++i) {
            mv[m_][slot + i]  = lv[i];
            mi_[m_][slot + i] = li[i];
        }
    }
    __syncthreads();
    if (tid < 16) {
        int row = tid;
        size_t obase = ((size_t)(qb + row) * PPARTS + part) * TOPK_;
        for (int t = 0; t < TOPK_; ++t) {
            float bv = 3.0e38f; int bi = 0x7fffffff; int bp = 0;
            for (int s = 0; s < 128; ++s) {
                float v = mv[row][s]; int ix = mi_[row][s];
                if (v < bv || (v == bv && ix < bi)) { bv = v; bi = ix; bp = s; }
            }
            pv[obase + t]  = bv;
            pix[obase + t] = bi;
            mv[row][bp] = 3.0e38f;
        }
    }
}

// ---------------------------------------------------------------------------
// Kernel 5: final top-16 of the 64*16 partition candidates per row, ascending,
// ties broken by lower index (mimics jax.lax.top_k on -dists).
// ---------------------------------------------------------------------------
__global__ __launch_bounds__(256) void kTop(const float* __restrict__ pv,
                                            const int* __restrict__ pix,
                                            float* __restrict__ outv,
                                            int* __restrict__ outi) {
    __shared__ float cv[PPARTS * TOPK_];
    __shared__ int   ci[PPARTS * TOPK_];
    __shared__ float rv[256];
    __shared__ int   ri[256];
    __shared__ int   rp[256];
    int b = blockIdx.x, tid = threadIdx.x;
    size_t base = (size_t)b * (PPARTS * TOPK_);
    for (int i = tid; i < PPARTS * TOPK_; i += 256) {
        cv[i] = pv[base + i];
        ci[i] = pix[base + i];
    }
    __syncthreads();
    for (int t = 0; t < TOPK_; ++t) {
        float bv = 3.0e38f; int bi = 0x7fffffff; int bp = 0;
        #pragma unroll
        for (int j = 0; j < 4; ++j) {
            int s = tid * 4 + j;
            float v = cv[s]; int ix = ci[s];
            if (v < bv || (v == bv && ix < bi)) { bv = v; bi = ix; bp = s; }
        }
        rv[tid] = bv; ri[tid] = bi; rp[tid] = bp;
        __syncthreads();
        for (int off = 128; off > 0; off >>= 1) {
            if (tid < off) {
                float v2 = rv[tid + off]; int i2 = ri[tid + off];
                if (v2 < rv[tid] || (v2 == rv[tid] && i2 < ri[tid])) {
                    rv[tid] = v2; ri[tid] = i2; rp[tid] = rp[tid + off];
                }
            }
            __syncthreads();
        }
        if (tid == 0) {
            outv[b * TOPK_ + t] = rv[0];
            outi[b * TOPK_ + t] = ri[0];
            cv[rp[0]] = 3.0e38f;
        }
        __syncthreads();
    }
}

// ---------------------------------------------------------------------------
// Host-side launch
// ---------------------------------------------------------------------------
extern "C" void kernel_launch(void* const* d_in, const int* in_sizes, int n_in,
                              void* d_out, int out_size, void* d_ws, size_t ws_size,
                              hipStream_t stream) {
    const float* query  = (const float*)d_in[0];   // (B, DM)
    const float* stored = (const float*)d_in[1];   // (N, DC)
    const float* resil  = (const float*)d_in[2];   // (N,)
    const float* Wp     = (const float*)d_in[3];   // (DM, DC)
    const float* Mm     = (const float*)d_in[4];   // (DC, DC)
    // d_in[5] = top_k scalar (compile-time TOPK_=16)

    // workspace layout (floats): G | qG | q2 | s2 | rinv | pvals | pidx (~9.3 MB)
    float* ws     = (float*)d_ws;
    float* wsG    = ws;                              // 64*64
    float* wsQG   = wsG   + DC_ * DC_;               // B*64
    float* wsQ2   = wsQG  + (size_t)B_ * DC_;        // B
    float* wsS2   = wsQ2  + B_;                      // N
    float* wsRinv = wsS2  + N_;                      // N
    float* wsPV   = wsRinv + N_;                     // B*PPARTS*16
    int*   wsPI   = (int*)(wsPV + (size_t)B_ * PPARTS * TOPK_);

    float* outv = (float*)d_out;                     // (B,16) fp32 values
    int*   outi = (int*)((float*)d_out + (size_t)B_ * TOPK_);  // (B,16) int32

    kMetric<<<1, 256, 0, stream>>>(Mm, wsG);
    kQuery <<<B_, 256, 0, stream>>>(query, Wp, wsG, wsQG, wsQ2);
    kS2    <<<256, 256, 0, stream>>>(stored, wsG, resil, wsS2, wsRinv);
    kMain  <<<dim3(PPARTS, B_ / 16), 128, 0, stream>>>(wsQG, wsQ2, stored,
                                                       wsS2, wsRinv, wsPV, wsPI);
    kTop   <<<B_, 256, 0, stream>>>(wsPV, wsPI, outv, outi);
}